// GraphConvolution_33749853012160
// MI455X (gfx1250) — compile-verified
//
#include <hip/hip_runtime.h>
#include <hip/hip_bf16.h>

// ---------------------------------------------------------------------------
// GCN layer: out = segment_sum(val * (X@W)[src], dst) + bias
//   N = 50000 nodes, E = 800000 edges, D_IN = D_OUT = 128
// Strategy (MI455X, wave32, WMMA):
//   K1: out[n][:] = bias            (so SpMM just accumulates)
//   K2: support = X @ W  via V_WMMA_F32_16X16X4_F32.
//       A tile (16x128 f32, 8KB) staged into LDS ONCE per block with
//       global_load_async_to_lds_b128 (ASYNCcnt) -> removes the 8x redundant
//       strided A reads that all 8 waves otherwise issue. LDS rows padded to
//       132 floats (132 % 64 == 4) so the ds_load_b64 A-fragment reads are
//       bank-conflict free across all 32 lanes.
//   K3: SpMM exploiting sorted dst: register-accumulate runs, flush with
//       global_atomic_add_f32 on dst change. support lives in d_ws
//       (25.6 MB -> resident in 192 MB L2, so the edge gather is L2 traffic).
// ---------------------------------------------------------------------------

#define N_NODES 50000
#define D_IN    128
#define D_OUT   128
#define LDS_PAD 132   // padded row stride (floats) for conflict-free LDS reads

typedef float v2f __attribute__((ext_vector_type(2)));
typedef float v8f __attribute__((ext_vector_type(8)));

// --- K1: out[n][d] = bias[d], float4-vectorized ----------------------------
__global__ __launch_bounds__(256) void gcn_init_bias(
    const float* __restrict__ bias, float* __restrict__ out) {
  int idx = blockIdx.x * 256 + threadIdx.x;          // over N*D/4 float4s
  float4 b = ((const float4*)bias)[idx & (D_OUT / 4 - 1)];
  ((float4*)out)[idx] = b;
}

// --- K2: support = X @ W with fp32 WMMA ------------------------------------
// One wave = one 16x16 tile of support. 8 waves/block cover the 8 N-tiles of
// one M-tile. Grid.x = N_NODES/16 = 3125 (exact), EXEC is all-ones as WMMA
// requires.
__global__ __launch_bounds__(256) void gcn_gemm_wmma(
    const float* __restrict__ X,   // [N, 128] row-major
    const float* __restrict__ W,   // [128, 128] row-major
    float* __restrict__ S) {       // [N, 128] row-major
  __shared__ float sA[16 * LDS_PAD];            // 8448 B

  const int tid    = threadIdx.x;
  const int lane   = tid & 31;
  const int tile_n = tid >> 5;                  // wave id 0..7
  const int tile_m = blockIdx.x;                // 0..3124

  // --- cooperatively stage the A tile (16 rows x 512B) via async-to-LDS ---
  // 256 threads: thread = (row, seg), each moves 32B = 2 x b128.
  // The 24-bit inst offset is added to BOTH the LDS and global addresses
  // (CDNA5 ISA), so one address setup covers both halves.
  {
    const int row = tid >> 4;                   // 0..15
    const int seg = tid & 15;                   // 0..15
    const float* g = X + (size_t)(tile_m * 16 + row) * D_IN + seg * 8;
    unsigned lds = (unsigned)(size_t)(&sA[row * LDS_PAD + seg * 8]);
    asm volatile(
        "global_load_async_to_lds_b128 %0, %1, off\n\t"
        "global_load_async_to_lds_b128 %0, %1, off offset:16"
        :: "v"(lds), "v"(g) : "memory");
    asm volatile("s_wait_asynccnt 0x0" ::: "memory");
  }
  __syncthreads();

  const int half = lane >> 4;                   // 0: lanes 0-15, 1: lanes 16-31
  const int l16  = lane & 15;
  const int koff = half * 2;                    // A/B VGPR k-offset (ISA layout)

  const int ncol = tile_n * 16 + l16;           // B/D col for this lane
  const float* aRow = &sA[l16 * LDS_PAD + koff];            // LDS, ds_load_b64
  const float* bcol = W + (size_t)koff * D_OUT + ncol;

  v8f c = {};
#pragma unroll
  for (int kb = 0; kb < D_IN; kb += 4) {
    // A 16x4: v0 = A[m][kb+koff], v1 = A[m][kb+koff+1]  (from LDS)
    v2f a = *(const v2f*)(aRow + kb);
    // B 4x16: v0 = W[kb+koff][n], v1 = W[kb+koff+1][n]
    v2f b;
    b.x = bcol[(size_t)kb * D_OUT];
    b.y = bcol[(size_t)kb * D_OUT + D_OUT];
    c = __builtin_amdgcn_wmma_f32_16x16x4_f32(
        /*neg_a=*/false, a, /*neg_b=*/false, b,
        /*c_mod=*/(short)0, c, /*reuse_a=*/false, /*reuse_b=*/false);
  }

  // D 16x16 f32: VGPR r -> row (r + 8*half), col = ncol
  const int mbase = tile_m * 16 + half * 8;
  float* outp = S + (size_t)mbase * D_OUT + ncol;
#pragma unroll
  for (int r = 0; r < 8; ++r) outp[(size_t)r * D_OUT] = c[r];
}

// --- K3: SpMM with run-length compressed atomics ---------------------------
// 128 threads/block: thread t owns feature dim t. Block processes a chunk of
// 128 consecutive edges. dst is sorted, so identical destinations form runs;
// accumulate a run in a register, flush via atomicAdd on change / chunk end.
#define EDGES_PER_BLOCK 128
__global__ __launch_bounds__(128) void gcn_spmm(
    const float* __restrict__ S,    // support [N, 128]
    const int*   __restrict__ src,
    const int*   __restrict__ dst,
    const float* __restrict__ val,
    float*       __restrict__ out,  // pre-initialized with bias
    int E) {
  const int d  = threadIdx.x;
  const int e0 = blockIdx.x * EDGES_PER_BLOCK;
  const int e1 = min(e0 + EDGES_PER_BLOCK, E);
  if (e0 >= E) return;

  float acc = 0.0f;
  int cur = dst[e0];                      // uniform -> scalar load
  for (int e = e0; e < e1; ++e) {
    int de = dst[e];
    if (de != cur) {
      atomicAdd(out + (size_t)cur * D_OUT + d, acc);
      acc = 0.0f;
      cur = de;
    }
    float v = val[e];
    acc = fmaf(v, S[(size_t)src[e] * D_OUT + d], acc);
  }
  atomicAdd(out + (size_t)cur * D_OUT + d, acc);
}

extern "C" void kernel_launch(void* const* d_in, const int* in_sizes, int n_in,
                              void* d_out, int out_size, void* d_ws, size_t ws_size,
                              hipStream_t stream) {
  const float* x       = (const float*)d_in[0];
  const int*   adj_src = (const int*)d_in[1];
  const int*   adj_dst = (const int*)d_in[2];
  const float* adj_val = (const float*)d_in[3];
  const float* weight  = (const float*)d_in[4];
  const float* bias    = (const float*)d_in[5];
  float*       out     = (float*)d_out;
  float*       support = (float*)d_ws;           // N_NODES * D_OUT floats

  const int E = in_sizes[1];

  // K1: out = bias (broadcast), N*D/4 float4 stores
  const int init_blocks = (N_NODES * D_OUT / 4) / 256;   // 6250, exact
  gcn_init_bias<<<init_blocks, 256, 0, stream>>>(bias, out);

  // K2: support = X @ W (WMMA fp32, async-LDS staged A)
  gcn_gemm_wmma<<<N_NODES / 16, 256, 0, stream>>>(x, weight, support);

  // K3: scatter-accumulate into out
  const int spmm_blocks = (E + EDGES_PER_BLOCK - 1) / EDGES_PER_BLOCK;
  gcn_spmm<<<spmm_blocks, 128, 0, stream>>>(support, adj_src, adj_dst, adj_val, out, E);
}